// IterativeUpdatingWorkingMemory_28484223107680
// MI455X (gfx1250) — compile-verified
//
#include <hip/hip_runtime.h>
#include <math.h>

// Shapes / constants from the reference
#define BB 16
#define KK 64
#define DD 128
#define NN 32768
#define TAU 0.8f
#define KEEP_N 48
#define ADD_N 16
#define RSQRT_D 0.08838834764831845f  // 1/sqrt(128)

typedef __attribute__((ext_vector_type(2))) float v2f;
typedef __attribute__((ext_vector_type(4))) float f4;
typedef __attribute__((ext_vector_type(8))) float v8f;

__device__ __forceinline__ float gelu_exact(float v) {
    // jax.nn.gelu(approximate=False): x * 0.5 * (1 + erf(x/sqrt(2)))
    return 0.5f * v * (1.0f + erff(v * 0.70710678118654752f));
}

// One wave computes a 16x16 f32 tile of  A[16 x Kdim] (LDS, row-major, ld=lda)
// times B[Kdim x 128] (global, row-major), columns [ncol0, ncol0+16).
// Uses V_WMMA_F32_16X16X4_F32 with ISA 7.12.2 f32 fragment layout:
//   A: lane<16 -> M=lane, K=k0+{0,1}; lane>=16 -> M=lane-16, K=k0+{2,3}
//   B: lane<16 -> N=lane, K=k0+{0,1}; lane>=16 -> N=lane-16, K=k0+{2,3}
__device__ __forceinline__ v8f wmma_tile_f32(const float* __restrict__ A, int lda,
                                             const float* __restrict__ Bmat,
                                             int Kdim, int ncol0) {
    const int lane = threadIdx.x & 31;
    const int mn   = lane & 15;        // M for A-frag, N for B-frag
    const int kh   = (lane >> 4) * 2;  // K sub-offset for the lane half
    v8f acc = {};
    for (int k0 = 0; k0 < Kdim; k0 += 4) {
        v2f a, b;
        a.x = A[mn * lda + k0 + kh + 0];
        a.y = A[mn * lda + k0 + kh + 1];
        b.x = Bmat[(k0 + kh + 0) * 128 + ncol0 + mn];
        b.y = Bmat[(k0 + kh + 1) * 128 + ncol0 + mn];
        acc = __builtin_amdgcn_wmma_f32_16x16x4_f32(
            false, a, false, b, (short)0, acc, false, false);
    }
    return acc;
}

// ---------------------------------------------------------------- mean over K
__global__ __launch_bounds__(128) void mean_rows_kernel(
    const float* __restrict__ X, float* __restrict__ out) {
    const int b = blockIdx.x, d = threadIdx.x;
    const float* p = X + (size_t)b * KK * DD + d;
    float s = 0.f;
    for (int k = 0; k < KK; ++k) s += p[k * DD];
    out[b * DD + d] = s * (1.0f / KK);
}

// ------------------------------------------------- pool MLP: [16,128] -> [16,128]
__global__ __launch_bounds__(256) void pool_kernel(
    const float* __restrict__ x, const float* __restrict__ lng,
    const float* __restrict__ lnb, const float* __restrict__ w1,
    const float* __restrict__ b1, const float* __restrict__ w2,
    const float* __restrict__ b2, float* __restrict__ q) {
    __shared__ float xn[16 * 128];
    __shared__ float hmid[16 * 128];
    __shared__ float mu[16], rstd[16];
    const int tid = threadIdx.x;

    for (int i = tid; i < 16 * 128; i += 256) xn[i] = x[i];
    __syncthreads();
    if (tid < 16) {
        float s = 0.f, ss = 0.f;
        for (int c = 0; c < 128; ++c) { float v = xn[tid * 128 + c]; s += v; ss += v * v; }
        float m = s * (1.0f / 128.0f);
        mu[tid] = m;
        rstd[tid] = rsqrtf(fmaxf(ss * (1.0f / 128.0f) - m * m, 0.f) + 1e-5f);
    }
    __syncthreads();
    for (int i = tid; i < 16 * 128; i += 256) {
        int r = i >> 7, c = i & 127;
        xn[i] = (xn[i] - mu[r]) * rstd[r] * lng[c] + lnb[c];
    }
    __syncthreads();

    const int w = tid >> 5, ncol0 = w * 16;
    const int lane = tid & 31, nc = ncol0 + (lane & 15), mh = (lane >> 4) * 8;

    v8f acc = wmma_tile_f32(xn, 128, w1, 128, ncol0);
    for (int r = 0; r < 8; ++r)
        hmid[(mh + r) * 128 + nc] = gelu_exact(acc[r] + b1[nc]);
    __syncthreads();

    acc = wmma_tile_f32(hmid, 128, w2, 128, ncol0);
    for (int r = 0; r < 8; ++r)
        q[(mh + r) * 128 + nc] = acc[r] + b2[nc];
}

// ---------------- shift MLP: rows = [W | q_b] (1024 x 256) -> delta; Weff = W+delta
__global__ __launch_bounds__(256) void shift_kernel(
    const float* __restrict__ W, const float* __restrict__ q,
    const float* __restrict__ lng, const float* __restrict__ lnb,
    const float* __restrict__ w1, const float* __restrict__ b1,
    const float* __restrict__ w2, const float* __restrict__ b2,
    float* __restrict__ Weff) {
    __shared__ float xn[16 * 256];
    __shared__ float hmid[16 * 128];
    __shared__ float mu[16], rstd[16];
    const int tid = threadIdx.x;
    const int r0 = blockIdx.x * 16;

    for (int i = tid; i < 16 * 256; i += 256) {
        int m = i >> 8, c = i & 255;
        int gr = r0 + m, b = gr >> 6;
        xn[i] = (c < 128) ? W[(size_t)gr * 128 + c] : q[b * 128 + (c - 128)];
    }
    __syncthreads();
    if (tid < 16) {
        float s = 0.f, ss = 0.f;
        for (int c = 0; c < 256; ++c) { float v = xn[tid * 256 + c]; s += v; ss += v * v; }
        float m = s * (1.0f / 256.0f);
        mu[tid] = m;
        rstd[tid] = rsqrtf(fmaxf(ss * (1.0f / 256.0f) - m * m, 0.f) + 1e-5f);
    }
    __syncthreads();
    for (int i = tid; i < 16 * 256; i += 256) {
        int m = i >> 8, c = i & 255;
        xn[i] = (xn[i] - mu[m]) * rstd[m] * lng[c] + lnb[c];
    }
    __syncthreads();

    const int w = tid >> 5, ncol0 = w * 16;
    const int lane = tid & 31, nc = ncol0 + (lane & 15), mh = (lane >> 4) * 8;

    v8f acc = wmma_tile_f32(xn, 256, w1, 256, ncol0);
    for (int r = 0; r < 8; ++r)
        hmid[(mh + r) * 128 + nc] = gelu_exact(acc[r] + b1[nc]);
    __syncthreads();

    acc = wmma_tile_f32(hmid, 128, w2, 128, ncol0);
    for (int r = 0; r < 8; ++r) {
        int gr = r0 + mh + r;
        Weff[(size_t)gr * 128 + nc] = W[(size_t)gr * 128 + nc] + acc[r] + b2[nc];
    }
}

// ----------- keep MLP: rows = [Weff | q_b] (1024 x 256) -> LN -> 256x128 -> gelu -> dot 128
__global__ __launch_bounds__(256) void keep_mlp_kernel(
    const float* __restrict__ Weff, const float* __restrict__ q,
    const float* __restrict__ lng, const float* __restrict__ lnb,
    const float* __restrict__ w1, const float* __restrict__ b1,
    const float* __restrict__ kw2, const float* __restrict__ kb2,
    float* __restrict__ klogit) {
    __shared__ float xn[16 * 256];
    __shared__ float hmid[16 * 128];
    __shared__ float mu[16], rstd[16];
    const int tid = threadIdx.x;
    const int r0 = blockIdx.x * 16;

    for (int i = tid; i < 16 * 256; i += 256) {
        int m = i >> 8, c = i & 255;
        int gr = r0 + m, b = gr >> 6;
        xn[i] = (c < 128) ? Weff[(size_t)gr * 128 + c] : q[b * 128 + (c - 128)];
    }
    __syncthreads();
    if (tid < 16) {
        float s = 0.f, ss = 0.f;
        for (int c = 0; c < 256; ++c) { float v = xn[tid * 256 + c]; s += v; ss += v * v; }
        float m = s * (1.0f / 256.0f);
        mu[tid] = m;
        rstd[tid] = rsqrtf(fmaxf(ss * (1.0f / 256.0f) - m * m, 0.f) + 1e-5f);
    }
    __syncthreads();
    for (int i = tid; i < 16 * 256; i += 256) {
        int m = i >> 8, c = i & 255;
        xn[i] = (xn[i] - mu[m]) * rstd[m] * lng[c] + lnb[c];
    }
    __syncthreads();

    const int w = tid >> 5, ncol0 = w * 16;
    const int lane = tid & 31, nc = ncol0 + (lane & 15), mh = (lane >> 4) * 8;

    v8f acc = wmma_tile_f32(xn, 256, w1, 256, ncol0);
    for (int r = 0; r < 8; ++r)
        hmid[(mh + r) * 128 + nc] = gelu_exact(acc[r] + b1[nc]);
    __syncthreads();

    if (tid < 16) {
        float s = 0.f;
        for (int c = 0; c < 128; ++c) s += hmid[tid * 128 + c] * kw2[c];
        klogit[r0 + tid] = s + kb2[0];
    }
}

// --------- keep: gumbel softmax over K=64, top-48 (descending), gather W rows
__global__ __launch_bounds__(64) void keep_select_kernel(
    const float* __restrict__ W, const float* __restrict__ klogit,
    const float* __restrict__ u_keep, float* __restrict__ keep_probs_out,
    float* __restrict__ Wnext) {
    __shared__ float p[64];
    __shared__ float rv[64];
    __shared__ int ri[64];
    __shared__ float red[64];
    const int b = blockIdx.x, t = threadIdx.x;

    float lg = klogit[b * 64 + t];
    float u = u_keep[b * 64 + t];
    float z = (lg + (-logf(-logf(u)))) * (1.0f / TAU);

    red[t] = z; __syncthreads();
    for (int s = 32; s > 0; s >>= 1) { if (t < s) red[t] = fmaxf(red[t], red[t + s]); __syncthreads(); }
    float mx = red[0]; __syncthreads();
    float e = expf(z - mx);
    red[t] = e; __syncthreads();
    for (int s = 32; s > 0; s >>= 1) { if (t < s) red[t] += red[t + s]; __syncthreads(); }
    float prob = e / red[0];
    keep_probs_out[b * 64 + t] = prob;
    p[t] = prob;
    __syncthreads();

    for (int j = 0; j < KEEP_N; ++j) {
        rv[t] = p[t]; ri[t] = t; __syncthreads();
        for (int s = 32; s > 0; s >>= 1) {
            if (t < s) {
                if (rv[t + s] > rv[t] || (rv[t + s] == rv[t] && ri[t + s] < ri[t])) {
                    rv[t] = rv[t + s]; ri[t] = ri[t + s];
                }
            }
            __syncthreads();
        }
        int sel = ri[0];
        const float* src = W + ((size_t)b * 64 + sel) * 128;
        float* dst = Wnext + ((size_t)b * 64 + j) * 128;
        dst[t] = src[t];
        dst[t + 64] = src[t + 64];
        if (t == 0) p[sel] = -__builtin_inff();
        __syncthreads();
    }
}

// --------- streaming GEMV over C (256 MB): one wave per row, NT loads, fused gumbel
__global__ __launch_bounds__(256) void logits_kernel(
    const float* __restrict__ C, const float* __restrict__ q,
    const float* __restrict__ h, const float* __restrict__ u_sel,
    float* __restrict__ z) {
    const int lane = threadIdx.x & 31;
    const int gw = blockIdx.x * 8 + (threadIdx.x >> 5);
    const int nw = gridDim.x * 8;
    const int totalRows = BB * NN;
    for (int row = gw; row < totalRows; row += nw) {
        const int b = row >> 15;  // N = 32768
        f4 c = __builtin_nontemporal_load((const f4*)(C + (size_t)row * 128) + lane);
        f4 qv = *((const f4*)(q + b * 128) + lane);
        float partial = c.x * qv.x + c.y * qv.y + c.z * qv.z + c.w * qv.w;
        for (int off = 16; off > 0; off >>= 1) partial += __shfl_xor(partial, off);
        if (lane == 0) {
            float logit = partial * RSQRT_D - h[row];  // INHIB = 1.0
            float g = -logf(-logf(u_sel[row]));
            z[row] = (logit + g) * (1.0f / TAU);
        }
    }
}

// --------- per-batch: softmax over N=32768 + top-16 (descending) + gather C rows
__global__ __launch_bounds__(1024) void select_kernel(
    const float* __restrict__ C, const float* __restrict__ z,
    float* __restrict__ sel_probs, float* __restrict__ Wnext) {
    __shared__ float sv[1024];
    __shared__ int si[1024];
    const int b = blockIdx.x, t = threadIdx.x;
    const float* zb = z + (size_t)b * NN;

    float vals[32];
    float mx = -__builtin_inff();
    for (int i = 0; i < 32; ++i) { vals[i] = zb[i * 1024 + t]; mx = fmaxf(mx, vals[i]); }
    sv[t] = mx; __syncthreads();
    for (int s = 512; s > 0; s >>= 1) { if (t < s) sv[t] = fmaxf(sv[t], sv[t + s]); __syncthreads(); }
    mx = sv[0]; __syncthreads();

    float se = 0.f;
    for (int i = 0; i < 32; ++i) se += expf(vals[i] - mx);
    sv[t] = se; __syncthreads();
    for (int s = 512; s > 0; s >>= 1) { if (t < s) sv[t] += sv[t + s]; __syncthreads(); }
    const float inv = 1.0f / sv[0];
    __syncthreads();

    float* pb = sel_probs + (size_t)b * NN;
    for (int i = 0; i < 32; ++i) pb[i * 1024 + t] = expf(vals[i] - mx) * inv;

    for (int j = 0; j < ADD_N; ++j) {
        float bv = -__builtin_inff();
        int bi = 0x7fffffff;
        for (int i = 0; i < 32; ++i) {
            int gi = i * 1024 + t;
            if (vals[i] > bv || (vals[i] == bv && gi < bi)) { bv = vals[i]; bi = gi; }
        }
        sv[t] = bv; si[t] = bi; __syncthreads();
        for (int s = 512; s > 0; s >>= 1) {
            if (t < s) {
                if (sv[t + s] > sv[t] || (sv[t + s] == sv[t] && si[t + s] < si[t])) {
                    sv[t] = sv[t + s]; si[t] = si[t + s];
                }
            }
            __syncthreads();
        }
        const int sel = si[0];
        if ((sel & 1023) == t) vals[sel >> 10] = -__builtin_inff();
        if (t < 128)
            Wnext[((size_t)b * 64 + 48 + j) * 128 + t] = C[((size_t)b * NN + sel) * 128 + t];
        __syncthreads();
    }
}

extern "C" void kernel_launch(void* const* d_in, const int* in_sizes, int n_in,
                              void* d_out, int out_size, void* d_ws, size_t ws_size,
                              hipStream_t stream) {
    const float* W        = (const float*)d_in[0];
    const float* C        = (const float*)d_in[1];
    const float* h        = (const float*)d_in[2];
    const float* u_keep   = (const float*)d_in[3];
    const float* u_sel    = (const float*)d_in[4];
    const float* pool_lng = (const float*)d_in[5];
    const float* pool_lnb = (const float*)d_in[6];
    const float* pool_w1  = (const float*)d_in[7];
    const float* pool_b1  = (const float*)d_in[8];
    const float* pool_w2  = (const float*)d_in[9];
    const float* pool_b2  = (const float*)d_in[10];
    const float* shift_lng = (const float*)d_in[11];
    const float* shift_lnb = (const float*)d_in[12];
    const float* shift_w1  = (const float*)d_in[13];
    const float* shift_b1  = (const float*)d_in[14];
    const float* shift_w2  = (const float*)d_in[15];
    const float* shift_b2  = (const float*)d_in[16];
    const float* keep_lng  = (const float*)d_in[17];
    const float* keep_lnb  = (const float*)d_in[18];
    const float* keep_w1   = (const float*)d_in[19];
    const float* keep_b1   = (const float*)d_in[20];
    const float* keep_w2   = (const float*)d_in[21];
    const float* keep_b2   = (const float*)d_in[22];

    float* ws = (float*)d_ws;
    float* mW   = ws;                    // 2048
    float* q1   = mW + BB * DD;          // 2048
    float* q2   = q1 + BB * DD;          // 2048
    float* Weff = q2 + BB * DD;          // 131072
    float* klog = Weff + BB * KK * DD;   // 1024
    float* z    = klog + BB * KK;        // 524288

    float* outW  = (float*)d_out;                 // [16,64,128]
    float* outKP = outW + BB * KK * DD;           // [16,64]
    float* outSP = outKP + BB * KK;               // [16,32768]

    // q1 = pool(mean(W))
    mean_rows_kernel<<<BB, 128, 0, stream>>>(W, mW);
    pool_kernel<<<1, 256, 0, stream>>>(mW, pool_lng, pool_lnb, pool_w1, pool_b1,
                                       pool_w2, pool_b2, q1);
    // W_eff = W + shift([W|q1])
    shift_kernel<<<BB * KK / 16, 256, 0, stream>>>(W, q1, shift_lng, shift_lnb,
                                                   shift_w1, shift_b1, shift_w2,
                                                   shift_b2, Weff);
    // q2 = pool(mean(W_eff))
    mean_rows_kernel<<<BB, 128, 0, stream>>>(Weff, mW);
    pool_kernel<<<1, 256, 0, stream>>>(mW, pool_lng, pool_lnb, pool_w1, pool_b1,
                                       pool_w2, pool_b2, q2);
    // keep logits + keep gumbel-softmax + top-48 gather (rows 0..47 of W_next)
    keep_mlp_kernel<<<BB * KK / 16, 256, 0, stream>>>(Weff, q2, keep_lng, keep_lnb,
                                                      keep_w1, keep_b1, keep_w2,
                                                      keep_b2, klog);
    keep_select_kernel<<<BB, 64, 0, stream>>>(W, klog, u_keep, outKP, outW);
    // big streaming GEMV over C with fused gumbel
    logits_kernel<<<4096, 256, 0, stream>>>(C, q2, h, u_sel, z);
    // softmax over N + top-16 gather (rows 48..63 of W_next)
    select_kernel<<<BB, 1024, 0, stream>>>(C, z, outSP, outW);
}